// MultiheadAttentionLocal_17102559772675
// MI455X (gfx1250) — compile-verified
//
#include <hip/hip_runtime.h>
#include <hip/hip_bf16.h>

typedef __attribute__((ext_vector_type(2)))  float v2f;
typedef __attribute__((ext_vector_type(8)))  float v8f;
typedef __attribute__((ext_vector_type(4)))  unsigned int u32x4;
typedef __attribute__((ext_vector_type(4)))  int i32x4;
typedef __attribute__((ext_vector_type(8)))  int i32x8;
typedef __attribute__((address_space(3)))    float lds_f;

#define EMB   256
#define NHEAD 8
#define DHEAD 32
#define KNBR  16

// ---------------------------------------------------------------------------
// TDM: DMA one 16-row x 256-col fp32 tile of W (row-major, stride 256) into
// LDS at byte offset lds_off.  D# built per CDNA5 ISA:
//   group0: count=1 | lds_addr | global_addr[56:0] | type=2
//   group1: data_size=2 (4B), tensor_dim0=256, tensor_dim1=wrows,
//           tile_dim0=256, tile_dim1=16, tensor_dim0_stride=256
//   group2/3: zero (2-D tensor)
// Tracked with TENSORcnt.
// ---------------------------------------------------------------------------
__device__ __forceinline__ void tdm_load_w_tile(const float* W, int wrow0,
                                                int wrows_total,
                                                unsigned int lds_off)
{
  const unsigned long long ga =
      (unsigned long long)(uintptr_t)(W + (size_t)wrow0 * EMB);

  u32x4 g0;
  g0[0] = 1u;                                      // count=1 (valid), no gather
  g0[1] = lds_off;                                 // LDS byte address
  g0[2] = (unsigned int)(ga & 0xFFFFFFFFull);      // global_addr[31:0]
  g0[3] = ((unsigned int)(ga >> 32) & 0x01FFFFFFu) // global_addr[56:32]
          | 0x80000000u;                           // type=2 ("image")

  i32x8 g1;
  g1[0] = (int)(2u << 16);                         // data_size = 4 bytes
  g1[1] = (int)((unsigned)EMB << 16);              // tensor_dim0 = 256 (bits 79:48)
  g1[2] = (int)((unsigned)wrows_total << 16);      // tensor_dim1 (bits 111:80)
  g1[3] = (int)((unsigned)EMB << 16);              // tile_dim0 = 256 (bits 127:112)
  g1[4] = 16;                                      // tile_dim1 = 16
  g1[5] = EMB;                                     // tensor_dim0_stride = 256
  g1[6] = 0;
  g1[7] = 0;                                       // tensor_dim1_stride unused (2-D)

  i32x4 g2 = {0, 0, 0, 0};
  i32x4 g3 = {0, 0, 0, 0};
#if __clang_major__ >= 23
  i32x8 g4 = {0, 0, 0, 0, 0, 0, 0, 0};
  __builtin_amdgcn_tensor_load_to_lds(g0, g1, g2, g3, g4, 0);
#else
  __builtin_amdgcn_tensor_load_to_lds(g0, g1, g2, g3, 0);
#endif
}

// ---------------------------------------------------------------------------
// One wave computes one 16x16 tile of  O[m0:, n0:] =
//   (X[m0:, :] @ Wtile^T + bias) * scale
// with A streamed from global and B read from the LDS-resident W tile.
//
// fp32 WMMA fragment layouts (CDNA5 ISA 7.12.2):
//   A 16x4 : lane l<16 -> (A[M=l][K=0], A[M=l][K=1]); lane l>=16 -> K=2,3
//   B 4x16 : mirrored, B[k][n] = W[n][k] -> 2 consecutive floats of a W row
//   C/D    : VGPR r, lane l -> C[m0 + r + (l>>4)*8][n0 + (l&15)]
// ---------------------------------------------------------------------------
__device__ __forceinline__ void gemm_tile_ldsB(
    const float* __restrict__ X, const float* wtile /* LDS, 16x256 */,
    const float* __restrict__ bias, float* __restrict__ O,
    int m0, int n0, float scale)
{
  const int lane  = threadIdx.x & 31;
  const int l15   = lane & 15;
  const int khalf = (lane >> 4) * 2;               // 0 or 2
  const float* arow = X + (size_t)(m0 + l15) * EMB + khalf;
  const float* brow = wtile + l15 * EMB + khalf;   // LDS row of W tile

  v8f acc = {};
#pragma unroll
  for (int k0 = 0; k0 < EMB; k0 += 4) {
    v2f a = *(const v2f*)(arow + k0);              // global_load_b64
    v2f b = *(const v2f*)(brow + k0);              // ds_load_b64
    acc = __builtin_amdgcn_wmma_f32_16x16x4_f32(
        false, a, false, b, (short)0, acc, false, false);
  }

  const float bv = bias[l15];
  const int rshift = (lane >> 4) * 8;
#pragma unroll
  for (int r = 0; r < 8; ++r)
    O[(size_t)(m0 + r + rshift) * EMB + n0 + l15] = (acc[r] + bv) * scale;
}

// ---------------------------------------------------------------------------
// Kernel 1: fused Q/K/V projection.  Block = 8 waves sharing one column tile
// (one 16-row slice of the packed 768x256 in_proj_weight, TDM-staged in LDS),
// covering 8 consecutive 16-row input tiles.  Q scaled by 1/sqrt(Dh).
// ---------------------------------------------------------------------------
__global__ void __launch_bounds__(256)
qkv_proj_kernel(const float* __restrict__ query, const float* __restrict__ keyb,
                const float* __restrict__ valb,  const float* __restrict__ Wp,
                const float* __restrict__ bp,
                float* __restrict__ qo, float* __restrict__ ko,
                float* __restrict__ vo, int mtiles)
{
  __shared__ float wtile[16 * EMB];                // 16 KB weight tile

  const int wv = threadIdx.x >> 5;
  const int tn = (int)blockIdx.x % 48;             // 48 column tiles over 768
  const int tm = ((int)blockIdx.x / 48) * 8 + wv;

  const int nglob = tn * 16;                       // packed weight row space
  const int which = nglob >> 8;                    // 0=q, 1=k, 2=v
  const int nloc  = nglob & 255;

  if (wv == 0) {
    const unsigned int lds_off =
        (unsigned int)(unsigned long long)(lds_f*)wtile;
    tdm_load_w_tile(Wp, nglob, 768, lds_off);
    __builtin_amdgcn_s_wait_tensorcnt(0);
  }
  __syncthreads();

  if (tm >= mtiles) return;

  const float* X = (which == 0) ? query : (which == 1) ? keyb : valb;
  float*       O = (which == 0) ? qo    : (which == 1) ? ko   : vo;
  const float scale = (which == 0) ? 0.17677669529663687f /* 1/sqrt(32) */ : 1.0f;

  gemm_tile_ldsB(X, wtile, bp + nglob, O, tm * 16, nloc, scale);
}

// ---------------------------------------------------------------------------
// Kernel 2: gathered attention.  One wave32 per query (unchanged; the k/v
// gather working set is 128 MB and stays resident in the 192 MB L2).
// ---------------------------------------------------------------------------
__global__ void __launch_bounds__(256)
attn_kernel(const float* __restrict__ qm, const float* __restrict__ km,
            const float* __restrict__ vm, const int* __restrict__ idx,
            float* __restrict__ attn_out, float* __restrict__ wsum, int tq)
{
  __shared__ float sw[8][KNBR * NHEAD];            // per-wave weights [nbr][head]

  const int wv   = threadIdx.x >> 5;
  const int lane = threadIdx.x & 31;
  const int q    = (int)blockIdx.x * 8 + wv;
  if (q >= tq) return;                             // wave-uniform

  const int nbr  = lane & 15;
  const int half = lane >> 4;

  const int  id    = idx[(size_t)q * KNBR + nbr];
  const bool valid = (id >= 0);
  const int  sid   = valid ? id : 0;

  const float* qrow = qm + (size_t)q   * EMB + half * 128;
  const float* krow = km + (size_t)sid * EMB + half * 128;

  // logits: 4 heads per lane, dot length 32 (q pre-scaled by 1/sqrt(Dh))
  float lg[4] = {0.f, 0.f, 0.f, 0.f};
#pragma unroll
  for (int t = 0; t < 4; ++t) {
#pragma unroll
    for (int d = 0; d < DHEAD; d += 4) {
      const float4 a = *(const float4*)(qrow + t * DHEAD + d);
      const float4 b = *(const float4*)(krow + t * DHEAD + d);
      lg[t] += a.x * b.x + a.y * b.y + a.z * b.z + a.w * b.w;
    }
    if (!valid) lg[t] = -__builtin_inff();
  }

  // masked softmax over 16 neighbors per head (shfl within 16-lane group)
  float wsum_part = 0.f;
#pragma unroll
  for (int t = 0; t < 4; ++t) {
    float m = lg[t];
#pragma unroll
    for (int o = 1; o <= 8; o <<= 1) m = fmaxf(m, __shfl_xor(m, o, 32));
    const float e = (m == -__builtin_inff()) ? 0.f : __expf(lg[t] - m);
    float s = e;
#pragma unroll
    for (int o = 1; o <= 8; o <<= 1) s += __shfl_xor(s, o, 32);
    const float w = (s > 0.f) ? (e / s) : 0.f;     // all-invalid -> zeros
    sw[wv][nbr * NHEAD + half * 4 + t] = w;
    wsum_part += w;
  }

  // weights.sum(axis=heads) / H
  wsum_part += __shfl_xor(wsum_part, 16, 32);
  if (half == 0)
    wsum[(size_t)q * KNBR + nbr] = wsum_part * (1.0f / NHEAD);

  __syncthreads();

  // weighted V accumulation: lane owns cols [lane*8, lane*8+8), single head
  const int h    = lane >> 2;
  const int eoff = lane * 8;
  const float* wrow2 = sw[wv];
  float acc[8] = {0.f, 0.f, 0.f, 0.f, 0.f, 0.f, 0.f, 0.f};
#pragma unroll
  for (int n = 0; n < KNBR; ++n) {
    const int   vid = __shfl(sid, n, 32);          // wave-uniform gather index
    const float wgt = wrow2[n * NHEAD + h];        // 0 for invalid neighbors
    const float* vrow = vm + (size_t)vid * EMB + eoff;
    const float4 a = *(const float4*)(vrow);
    const float4 b = *(const float4*)(vrow + 4);
    acc[0] += wgt * a.x; acc[1] += wgt * a.y; acc[2] += wgt * a.z; acc[3] += wgt * a.w;
    acc[4] += wgt * b.x; acc[5] += wgt * b.y; acc[6] += wgt * b.z; acc[7] += wgt * b.w;
  }

  float* orow = attn_out + (size_t)q * EMB + eoff;
  const float4 o1 = {acc[0], acc[1], acc[2], acc[3]};
  const float4 o2 = {acc[4], acc[5], acc[6], acc[7]};
  *(float4*)(orow)     = o1;
  *(float4*)(orow + 4) = o2;
}

// ---------------------------------------------------------------------------
// Kernel 3: output projection GEMM.  Same TDM-staged-B structure; 16 column
// tiles over the 256x256 out_proj_weight.
// ---------------------------------------------------------------------------
__global__ void __launch_bounds__(256)
out_proj_kernel(const float* __restrict__ A, const float* __restrict__ Wo,
                const float* __restrict__ bo, float* __restrict__ O, int mtiles)
{
  __shared__ float wtile[16 * EMB];

  const int wv = threadIdx.x >> 5;
  const int tn = (int)blockIdx.x & 15;             // 16 column tiles over 256
  const int tm = ((int)blockIdx.x >> 4) * 8 + wv;

  if (wv == 0) {
    const unsigned int lds_off =
        (unsigned int)(unsigned long long)(lds_f*)wtile;
    tdm_load_w_tile(Wo, tn * 16, 256, lds_off);
    __builtin_amdgcn_s_wait_tensorcnt(0);
  }
  __syncthreads();

  if (tm >= mtiles) return;
  gemm_tile_ldsB(A, wtile, bo + tn * 16, O, tm * 16, tn * 16, 1.0f);
}

// ---------------------------------------------------------------------------
extern "C" void kernel_launch(void* const* d_in, const int* in_sizes, int n_in,
                              void* d_out, int out_size, void* d_ws, size_t ws_size,
                              hipStream_t stream)
{
  const float* query = (const float*)d_in[0];
  const float* keyb  = (const float*)d_in[1];
  const float* valb  = (const float*)d_in[2];
  const int*   ip    = (const int*)  d_in[3];
  const float* Wp    = (const float*)d_in[4];   // [768,256] packed q/k/v
  const float* bp    = (const float*)d_in[5];   // [768]
  const float* Wo    = (const float*)d_in[6];   // [256,256]
  const float* bo    = (const float*)d_in[7];   // [256]

  const int tq = in_sizes[0] / EMB;             // 65536

  // workspace: q | k | v | attn  (each tq*256 f32)
  float* qbuf = (float*)d_ws;
  float* kbuf = qbuf + (size_t)tq * EMB;
  float* vbuf = kbuf + (size_t)tq * EMB;
  float* abuf = vbuf + (size_t)tq * EMB;

  float* out   = (float*)d_out;                 // [tq,256]
  float* wsumo = out + (size_t)tq * EMB;        // [tq,16]

  const int mtiles  = tq / 16;
  const int mblocks = (mtiles + 7) / 8;         // 8 row tiles per block

  qkv_proj_kernel<<<48 * mblocks, 256, 0, stream>>>(query, keyb, valb, Wp, bp,
                                                    qbuf, kbuf, vbuf, mtiles);

  attn_kernel<<<(tq + 7) / 8, 256, 0, stream>>>(qbuf, kbuf, vbuf, ip,
                                                abuf, wsumo, tq);

  out_proj_kernel<<<16 * mblocks, 256, 0, stream>>>(abuf, Wo, bo, out, mtiles);
}